// OccupancyGridSampler_63273458205200
// MI455X (gfx1250) — compile-verified
//
#include <hip/hip_runtime.h>

typedef __attribute__((ext_vector_type(2))) float v2f;
typedef __attribute__((ext_vector_type(8))) float v8f;

static constexpr int kS   = 256;   // samples per ray
static constexpr int kRes = 128;   // density grid resolution

__device__ __forceinline__ float fetch_density(const float* __restrict__ g,
                                               float ox, float oy, float oz,
                                               float dx, float dy, float dz,
                                               float t) {
    // position along ray
    float px = fmaf(dx, t, ox);
    float py = fmaf(dy, t, oy);
    float pz = fmaf(dz, t, oz);
    // nearest voxel: floor((p - (-1)) * 64), clamped to [0,127]
    int ix = (int)floorf((px + 1.0f) * 64.0f);
    int iy = (int)floorf((py + 1.0f) * 64.0f);
    int iz = (int)floorf((pz + 1.0f) * 64.0f);
    ix = min(max(ix, 0), kRes - 1);
    iy = min(max(iy, 0), kRes - 1);
    iz = min(max(iz, 0), kRes - 1);
    // regular-temporal load: keep the 8MB grid resident in the 192MB L2
    return g[((ix * kRes) + iy) * kRes + iz];
}

__global__ __launch_bounds__(256) void occ_sampler_wmma(
    const float* __restrict__ origins,
    const float* __restrict__ directions,
    const float* __restrict__ nears,
    const float* __restrict__ fars,
    const float* __restrict__ t_rand,
    const float* __restrict__ grid,
    float* __restrict__ out_w,
    float* __restrict__ out_m,
    int n_rays)
{
    const int tid  = blockIdx.x * blockDim.x + threadIdx.x;
    const int ray  = tid >> 5;          // one wave32 per ray (wave-uniform)
    const int lane = threadIdx.x & 31;
    const int row  = lane & 15;         // matrix row M this lane feeds (A operand)
    const int hi   = lane >> 4;         // lane half: selects K sub-pair
    if (ray >= n_rays) return;          // wave-uniform; grid sized exactly anyway

    const float ox = origins[ray * 3 + 0], oy = origins[ray * 3 + 1], oz = origins[ray * 3 + 2];
    const float dx = directions[ray * 3 + 0], dy = directions[ray * 3 + 1], dz = directions[ray * 3 + 2];
    const float nr = nears[ray];
    const float fr = fars[ray];
    const float h  = (fr - nr) * (1.0f / (float)kS);   // uniform bin width

    const float* tr = t_rand + (size_t)ray * kS;

    // Lane (row,hi) owns samples s(j,e) = 16*row + 4*j + 2*hi + e, j=0..3, e=0..1.
    float r0[4], r1[4];      // randoms of the two owned samples per j
    float den0[4], den1[4];  // grid densities
#pragma unroll
    for (int j = 0; j < 4; ++j) {
        const int s0 = 16 * row + 4 * j + 2 * hi;
        // read-once stream: non-temporal so it doesn't evict the grid from L2
        const v2f rr = __builtin_nontemporal_load((const v2f*)(tr + s0));
        r0[j] = rr.x; r1[j] = rr.y;
        // ts = near + (s + r) * h   (linspace bins + stratified jitter)
        const float ta = fmaf((float)s0 + rr.x, h, nr);
        const float tb = fmaf((float)(s0 + 1) + rr.y, h, nr);
        den0[j] = fetch_density(grid, ox, oy, oz, dx, dy, dz, ta);
        den1[j] = fetch_density(grid, ox, oy, oz, dx, dy, dz, tb);
    }

    // Neighbor randoms for deltas: delta_s = h*(1 + r_{s+1} - r_s); last: h*(1 - r_255).
    float pr[4];                                   // high-partner's r(j,0) -> used by low half
#pragma unroll
    for (int j = 0; j < 4; ++j) pr[j] = __shfl(r0[j], row + 16, 32);
    const float r00_next = __shfl(r0[0], (row + 1) & 15, 32);  // next row's first r -> used by high half

    // X(j,e) = delta * density, held in the native WMMA A-operand layout.
    float X0[4], X1[4];
#pragma unroll
    for (int j = 0; j < 4; ++j) {
        X0[j] = h * (1.0f + r1[j] - r0[j]) * den0[j];           // e=0: successor in-lane
        float rn = (hi == 0) ? pr[j]
                 : (j < 3)   ? r0[j + 1]
                             : r00_next;
        float d1 = h * (1.0f + rn - r1[j]);
        if (hi == 1 && j == 3 && row == 15)                      // global last sample s=255
            d1 = h * (1.0f - r1[j]);                             // fars - ts[-1]
        X1[j] = d1 * den1[j];
    }

    // ---- matrix prefix-scan via V_WMMA_F32_16X16X4_F32 ----
    // B operand (4x16 slice j), lane (n=row, hi): VGPR p holds B[4j + p + 2*hi][n].
    const int n = row;
    v8f accP = {};   // X * U'  (strict upper ones)  -> within-row exclusive prefix
    v8f accX = {};   // X * I                        -> X transposed into C/D layout
    v8f accR = {};   // X * J   (all ones)           -> row sums broadcast across N
#pragma unroll
    for (int j = 0; j < 4; ++j) {
        v2f A;  A.x = X0[j];  A.y = X1[j];
        const int k0 = 4 * j + 2 * hi;
        const int k1 = k0 + 1;
        v2f Bu; Bu.x = (k0 < n) ? 1.0f : 0.0f; Bu.y = (k1 < n) ? 1.0f : 0.0f;
        v2f Bi; Bi.x = (k0 == n) ? 1.0f : 0.0f; Bi.y = (k1 == n) ? 1.0f : 0.0f;
        v2f Bj; Bj.x = 1.0f; Bj.y = 1.0f;
        accP = __builtin_amdgcn_wmma_f32_16x16x4_f32(false, A, false, Bu, (short)0, accP, false, false);
        accX = __builtin_amdgcn_wmma_f32_16x16x4_f32(false, A, false, Bi, (short)0, accX, false, false);
        accR = __builtin_amdgcn_wmma_f32_16x16x4_f32(false, A, false, Bj, (short)0, accR, false, false);
    }

    // Exclusive scan over the 16 row sums: 8 in-register + one cross-half shuffle.
    float off[8];
    float run = 0.0f;
#pragma unroll
    for (int v = 0; v < 8; ++v) { off[v] = run; run += accR[v]; }
    const float tot_low = __shfl(run, n, 32);     // low-half total (rows 0..7)
    const float base = hi ? tot_low : 0.0f;

    // Epilogue in C/D layout: lane holds (M = v + 8*hi, N = n), sample s = 16*M + N.
    const size_t obase = (size_t)ray * kS;
#pragma unroll
    for (int v = 0; v < 8; ++v) {
        const float E     = accP[v] + off[v] + base;     // exclusive cumsum of delta*density
        const float T     = __expf(-E);                  // transmittance
        const float alpha = 1.0f - __expf(-accX[v]);
        const float w     = alpha * T;
        const int   M     = v + 8 * hi;
        const size_t o    = obase + (size_t)(16 * M + n);
        // write-once streams: non-temporal stores, keep L2 for the grid
        __builtin_nontemporal_store(w, &out_w[o]);
        __builtin_nontemporal_store((w >= 1e-4f) ? 1.0f : 0.0f, &out_m[o]);
    }
}

extern "C" void kernel_launch(void* const* d_in, const int* in_sizes, int n_in,
                              void* d_out, int out_size, void* d_ws, size_t ws_size,
                              hipStream_t stream) {
    const float* origins    = (const float*)d_in[0];
    const float* directions = (const float*)d_in[1];
    const float* nears      = (const float*)d_in[2];
    const float* fars       = (const float*)d_in[3];
    const float* t_rand     = (const float*)d_in[4];
    const float* grid       = (const float*)d_in[5];

    const int N = in_sizes[2];                 // number of rays (nears is [N])
    float* out_w = (float*)d_out;              // weights  [N, 256]
    float* out_m = out_w + (size_t)N * kS;     // mask     [N, 256] (1.0 / 0.0)

    const int threads = 256;                   // 8 waves -> 8 rays per block
    const int blocks  = (N * 32 + threads - 1) / threads;
    occ_sampler_wmma<<<blocks, threads, 0, stream>>>(
        origins, directions, nears, fars, t_rand, grid, out_w, out_m, N);
}